// treeLayer_489626271943
// MI455X (gfx1250) — compile-verified
//
#include <hip/hip_runtime.h>
#include <stdint.h>

// ---------------------------------------------------------------------------
// treeLayer: out[b,j] = leakyrelu( in[b,2j]*k[2j] + in[b,2j+1]*k[2j+1] + bias[j] )
// B=2048, N=32768, DIV=2. Memory-bound: 256MB in + 128MB out => ~16.5us floor
// at 23.3 TB/s. Strategy: wide NT b128 streams for in/out; kernel+bias tile
// staged once per block into LDS via CDNA5 async-tensor path and reused over
// ROWS_PER_BLOCK rows from registers.
// ---------------------------------------------------------------------------

#define B_ROWS 2048
#define N_IN   32768
#define N_OUT  (N_IN / 2)
#define ALPHA  0.01f

#define TPB 256
#define VEC 4
#define COLS_PER_BLOCK (TPB * VEC)   // 1024 output columns per block tile
#define ROWS_PER_BLOCK 8             // rows amortizing one LDS staging

// Native clang vector: required by __builtin_nontemporal_load/store
// (HIP's float4 is a class and is rejected). Same 16-byte layout/alignment.
typedef float v4f __attribute__((ext_vector_type(4)));

__global__ __launch_bounds__(TPB) void treeLayer_fused_kernel(
    const float* __restrict__ in,     // [B, N_IN]
    const float* __restrict__ kern,   // [N_IN]
    const float* __restrict__ bias,   // [N_OUT]
    float* __restrict__ out)          // [B, N_OUT]
{
    __shared__ float s_k[2 * COLS_PER_BLOCK];  // 8 KB: kernel[2*j0 .. 2*j0+2048)
    __shared__ float s_b[COLS_PER_BLOCK];      // 4 KB: bias[j0 .. j0+1024)

    const int tid     = threadIdx.x;
    const int j0      = blockIdx.x * COLS_PER_BLOCK;   // first output col of tile
    const int rowBase = blockIdx.y * ROWS_PER_BLOCK;

    // ---- Stage kernel + bias slices into LDS with gfx1250 async loads -----
    // GVS mode: mem = SGPR_base + VGPR_off + IOFFSET; per ISA 08 §4.4 the
    // IOFFSET is added to BOTH the global and the LDS address, so offset:4096
    // fetches the second half of the 8 KB kernel slice in one instruction.
    {
        unsigned ldsK = (unsigned)(uintptr_t)&s_k[tid * 4];
        unsigned gK   = (unsigned)((2 * j0 + tid * 4) * sizeof(float));
        asm volatile("global_load_async_to_lds_b128 %0, %1, %2 offset:0"
                     :: "v"(ldsK), "v"(gK), "s"(kern) : "memory");
        asm volatile("global_load_async_to_lds_b128 %0, %1, %2 offset:4096"
                     :: "v"(ldsK), "v"(gK), "s"(kern) : "memory");

        unsigned ldsB = (unsigned)(uintptr_t)&s_b[tid * 4];
        unsigned gB   = (unsigned)((j0 + tid * 4) * sizeof(float));
        asm volatile("global_load_async_to_lds_b128 %0, %1, %2 offset:0"
                     :: "v"(ldsB), "v"(gB), "s"(bias) : "memory");

        asm volatile("s_wait_asynccnt 0" ::: "memory");
    }
    __syncthreads();

    // Pull this thread's weights/bias into registers once; reused for 8 rows.
    const v4f* sk4 = reinterpret_cast<const v4f*>(s_k);
    const v4f* sb4 = reinterpret_cast<const v4f*>(s_b);
    const v4f k0 = sk4[2 * tid];
    const v4f k1 = sk4[2 * tid + 1];
    const v4f bb = sb4[tid];

    const v4f* in4  = reinterpret_cast<const v4f*>(in);
    v4f*       out4 = reinterpret_cast<v4f*>(out);

    const long inBase0  = (long)rowBase * (N_IN  / 4) + j0 / 2 + 2 * tid;
    const long outBase0 = (long)rowBase * (N_OUT / 4) + j0 / 4 + tid;

#pragma unroll
    for (int r = 0; r < ROWS_PER_BLOCK; ++r) {
        const long ib = inBase0 + (long)r * (N_IN / 4);
        // 256MB input is read exactly once: non-temporal so it doesn't evict
        // the reused kernel/bias lines from L2.
        v4f a0 = __builtin_nontemporal_load(&in4[ib]);
        v4f a1 = __builtin_nontemporal_load(&in4[ib + 1]);

        v4f o;
        o.x = fmaf(a0.x, k0.x, fmaf(a0.y, k0.y, bb.x));
        o.y = fmaf(a0.z, k0.z, fmaf(a0.w, k0.w, bb.y));
        o.z = fmaf(a1.x, k1.x, fmaf(a1.y, k1.y, bb.z));
        o.w = fmaf(a1.z, k1.z, fmaf(a1.w, k1.w, bb.w));

        // leaky_relu(x) == max(x, alpha*x) for alpha in (0,1): branchless.
        o.x = fmaxf(o.x, ALPHA * o.x);
        o.y = fmaxf(o.y, ALPHA * o.y);
        o.z = fmaxf(o.z, ALPHA * o.z);
        o.w = fmaxf(o.w, ALPHA * o.w);

        __builtin_nontemporal_store(o, &out4[outBase0 + (long)r * (N_OUT / 4)]);
    }
}

extern "C" void kernel_launch(void* const* d_in, const int* in_sizes, int n_in,
                              void* d_out, int out_size, void* d_ws, size_t ws_size,
                              hipStream_t stream) {
    const float* in   = (const float*)d_in[0];  // [B, N]
    const float* kern = (const float*)d_in[1];  // [1, N]
    const float* bias = (const float*)d_in[2];  // [N/2]
    float*       out  = (float*)d_out;          // [B, N/2]

    dim3 grid(N_OUT / COLS_PER_BLOCK,           // 16 column tiles
              B_ROWS / ROWS_PER_BLOCK);         // 256 row groups
    treeLayer_fused_kernel<<<grid, TPB, 0, stream>>>(in, kern, bias, out);
}